// CausalLinearAttention_90640989815099
// MI455X (gfx1250) — compile-verified
//
#include <hip/hip_runtime.h>
#include <hip/hip_bf16.h>

// Causal linear attention, chunk-parallel 3-pass formulation for MI455X (gfx1250).
// B=2 H=8 T=2048 D=64, fp32. Chunk C=64, NC=32 chunks, BH=16 -> 512 parallel WGs
// in passes 1 and 3. All matmuls use V_WMMA_F32_16X16X4_F32 (fp32 in/out).

#define B_  2
#define H_  8
#define T_  2048
#define D_  64
#define BH  16
#define C_  64
#define NC  32           // T_/C_
#define SSTR 4160        // per-chunk state: 64*64 S + 64 z floats
#define LD  68           // padded LDS row stride (floats) -> conflict-free column walks

typedef __attribute__((ext_vector_type(2))) float v2f;
typedef __attribute__((ext_vector_type(8))) float v8f;

static __device__ __forceinline__ v8f wmma4(v2f a, v2f b, v8f c) {
  // D(16x16,f32) = A(16x4,f32) * B(4x16,f32) + C
  return __builtin_amdgcn_wmma_f32_16x16x4_f32(
      false, a, false, b, (short)0, c, false, false);
}

static __device__ __forceinline__ float fmap(float x) {
  // elu(x)+1 : x>0 ? x+1 : exp(x)
  return x > 0.f ? x + 1.f : __expf(x);
}

// ---------------- Pass 1: per-chunk state S_c = K'^T V, z_c = sum K' ----------------
__global__ __launch_bounds__(128) void k_state(const float* __restrict__ k,
                                               const float* __restrict__ v,
                                               float* __restrict__ ws) {
  __shared__ float kc[C_][LD];
  __shared__ float vc[C_][LD];
  const int bh = blockIdx.x / NC, c = blockIdx.x % NC;
  const int tid = threadIdx.x;
  const float* kg = k + ((size_t)bh * T_ + (size_t)c * C_) * D_;
  const float* vg = v + ((size_t)bh * T_ + (size_t)c * C_) * D_;

  for (int i = tid; i < C_ * D_ / 4; i += 128) {   // 1024 float4s
    const int row = i >> 4, col = (i & 15) << 2;
    float4 kk4 = ((const float4*)kg)[i];
    kc[row][col + 0] = fmap(kk4.x); kc[row][col + 1] = fmap(kk4.y);
    kc[row][col + 2] = fmap(kk4.z); kc[row][col + 3] = fmap(kk4.w);
    float4 vv4 = ((const float4*)vg)[i];
    vc[row][col + 0] = vv4.x; vc[row][col + 1] = vv4.y;
    vc[row][col + 2] = vv4.z; vc[row][col + 3] = vv4.w;
  }
  __syncthreads();

  const int w = tid >> 5, lane = tid & 31, lh = lane >> 4, ln = lane & 15;
  float* sout = ws + (size_t)(bh * NC + c) * SSTR;

  // S[d][e] = sum_t K'[t][d] * V[t][e]; wave w owns cols e in [16w,16w+16)
  for (int mt = 0; mt < 4; ++mt) {
    v8f acc = {0.f, 0.f, 0.f, 0.f, 0.f, 0.f, 0.f, 0.f};
#pragma unroll
    for (int kk = 0; kk < 16; ++kk) {
      const int k0 = 4 * kk + 2 * lh;
      v2f a = {kc[k0][16 * mt + ln], kc[k0 + 1][16 * mt + ln]};  // A[m=d][k=t]
      v2f b = {vc[k0][16 * w + ln],  vc[k0 + 1][16 * w + ln]};   // B[k=t][n=e]
      acc = wmma4(a, b, acc);
    }
#pragma unroll
    for (int vv = 0; vv < 8; ++vv) {
      const int drow = 16 * mt + vv + 8 * lh;
      sout[drow * 64 + 16 * w + ln] = acc[vv];
    }
  }
  if (tid < 32) {  // z_c[d] = sum_t K'[t][d]
#pragma unroll
    for (int rep = 0; rep < 2; ++rep) {
      const int d = lane + 32 * rep;
      float s = 0.f;
      for (int t = 0; t < C_; ++t) s += kc[t][d];
      sout[4096 + d] = s;
    }
  }
}

// ---------------- Pass 2: exclusive prefix sum of states over chunks ----------------
__global__ __launch_bounds__(128) void k_scan(float* __restrict__ ws) {
  const int bh = blockIdx.x, tid = threadIdx.x;
  for (int e = tid; e < SSTR; e += 128) {
    const size_t base = (size_t)bh * NC * SSTR + e;
    float vals[NC];
#pragma unroll
    for (int c = 0; c < NC; ++c) vals[c] = ws[base + (size_t)c * SSTR];
    float run = 0.f;
#pragma unroll
    for (int c = 0; c < NC; ++c) {
      ws[base + (size_t)c * SSTR] = run;
      run += vals[c];
    }
  }
}

// ---------------- Pass 3: out = mask(Q'K'^T)V + Q'S_prev, / (den + eps) ----------------
__global__ __launch_bounds__(128) void k_out(const float* __restrict__ q,
                                             const float* __restrict__ k,
                                             const float* __restrict__ v,
                                             const float* __restrict__ ws,
                                             float* __restrict__ out) {
  __shared__ float qc[C_][LD];
  __shared__ float kc[C_][LD];
  __shared__ float vc[C_][LD];
  __shared__ float sP[64][LD];
  __shared__ float zP[64];
  __shared__ float abuf[4][16][LD];
  __shared__ float denb[4][16];

  const int bh = blockIdx.x / NC, c = blockIdx.x % NC;
  const int tid = threadIdx.x;
  const float* qg = q + ((size_t)bh * T_ + (size_t)c * C_) * D_;
  const float* kg = k + ((size_t)bh * T_ + (size_t)c * C_) * D_;
  const float* vg = v + ((size_t)bh * T_ + (size_t)c * C_) * D_;
  const float* sg = ws + (size_t)(bh * NC + c) * SSTR;

  for (int i = tid; i < C_ * D_ / 4; i += 128) {
    const int row = i >> 4, col = (i & 15) << 2;
    float4 q4 = ((const float4*)qg)[i];
    qc[row][col + 0] = fmap(q4.x); qc[row][col + 1] = fmap(q4.y);
    qc[row][col + 2] = fmap(q4.z); qc[row][col + 3] = fmap(q4.w);
    float4 k4 = ((const float4*)kg)[i];
    kc[row][col + 0] = fmap(k4.x); kc[row][col + 1] = fmap(k4.y);
    kc[row][col + 2] = fmap(k4.z); kc[row][col + 3] = fmap(k4.w);
    float4 v4 = ((const float4*)vg)[i];
    vc[row][col + 0] = v4.x; vc[row][col + 1] = v4.y;
    vc[row][col + 2] = v4.z; vc[row][col + 3] = v4.w;
    float4 s4 = ((const float4*)sg)[i];
    sP[row][col + 0] = s4.x; sP[row][col + 1] = s4.y;
    sP[row][col + 2] = s4.z; sP[row][col + 3] = s4.w;
  }
  if (tid < 64) zP[tid] = sg[4096 + tid];
  __syncthreads();

  const int w = tid >> 5, lane = tid & 31, lh = lane >> 4, ln = lane & 15;

  // Score tiles A[w-rows][nt-cols], only nt <= w survive the causal mask.
  for (int nt = 0; nt <= w; ++nt) {   // uniform per wave
    v8f acc = {0.f, 0.f, 0.f, 0.f, 0.f, 0.f, 0.f, 0.f};
#pragma unroll
    for (int kk = 0; kk < 16; ++kk) {
      const int k0 = 4 * kk + 2 * lh;
      v2f a = {qc[16 * w + ln][k0],  qc[16 * w + ln][k0 + 1]};   // A[m][k=d]
      v2f b = {kc[16 * nt + ln][k0], kc[16 * nt + ln][k0 + 1]};  // B[k=d][n] = K'^T
      acc = wmma4(a, b, acc);
    }
#pragma unroll
    for (int vv = 0; vv < 8; ++vv) {
      const int M = vv + 8 * lh;
      const float val = (nt < w || ln <= M) ? acc[vv] : 0.f;  // causal mask on diag tile
      abuf[w][M][16 * nt + ln] = val;
    }
  }

  // Denominator: rowsum(masked A) + q' . z_prev  (lanes 0-15, per-wave)
  if (lane < 16) {
    const int i = ln;
    float s = 0.f;
    const int lim = 16 * (w + 1);
    for (int cc = 0; cc < lim; ++cc) s += abuf[w][i][cc];
    for (int d = 0; d < 64; ++d) s += qc[16 * w + i][d] * zP[d];
    denb[w][i] = s;
  }

  // out = A_masked @ V  +  Q' @ S_prev
  for (int nt = 0; nt < 4; ++nt) {
    v8f o = {0.f, 0.f, 0.f, 0.f, 0.f, 0.f, 0.f, 0.f};
#pragma unroll
    for (int kk = 0; kk < 16; ++kk) {   // inter-chunk: K = D
      const int k0 = 4 * kk + 2 * lh;
      v2f a = {qc[16 * w + ln][k0], qc[16 * w + ln][k0 + 1]};
      v2f b = {sP[k0][16 * nt + ln], sP[k0 + 1][16 * nt + ln]};
      o = wmma4(a, b, o);
    }
    const int kkmax = 4 * (w + 1);      // A cols beyond the diagonal block are zero
    for (int kk = 0; kk < kkmax; ++kk) {  // intra-chunk: K = chunk rows
      const int k0 = 4 * kk + 2 * lh;
      v2f a = {abuf[w][ln][k0], abuf[w][ln][k0 + 1]};
      v2f b = {vc[k0][16 * nt + ln], vc[k0 + 1][16 * nt + ln]};
      o = wmma4(a, b, o);
    }
#pragma unroll
    for (int vv = 0; vv < 8; ++vv) {
      const int M = vv + 8 * lh;
      const float den = denb[w][M] + 1e-6f;
      const size_t row = (size_t)bh * T_ + (size_t)c * C_ + 16 * w + M;
      out[row * D_ + 16 * nt + ln] = o[vv] / den;
    }
  }
}

extern "C" void kernel_launch(void* const* d_in, const int* in_sizes, int n_in,
                              void* d_out, int out_size, void* d_ws, size_t ws_size,
                              hipStream_t stream) {
  const float* q = (const float*)d_in[0];
  const float* k = (const float*)d_in[1];
  const float* v = (const float*)d_in[2];
  float* out = (float*)d_out;
  float* ws = (float*)d_ws;   // needs BH*NC*SSTR*4 = 8.52 MB

  k_state<<<BH * NC, 128, 0, stream>>>(k, v, ws);
  k_scan<<<BH, 128, 0, stream>>>(ws);
  k_out<<<BH * NC, 128, 0, stream>>>(q, k, v, ws, out);
}